// UnifiedCoreFlow_25383256719437
// MI455X (gfx1250) — compile-verified
//
#include <hip/hip_runtime.h>

// Problem constants (match reference)
#define B_DIM 1024
#define N_DIM 1024
#define L_DIM 24
#define N_OUT 10

#define BLK_M 32            // rows gathered into LDS per block
#define LDS_STRIDE 1028     // 1024 + 4 pad -> stride mod 64 == 4 (bank-conflict-free)

typedef float v2f __attribute__((ext_vector_type(2)));
typedef float v8f __attribute__((ext_vector_type(8)));

// ---------------------------------------------------------------------------
// Fused per-layer kernel: gather rows into LDS, FP32-WMMA GEMM, quantize.
//   hout[b,o] = quant( clip( sum_i hin[b, idx[i]] * W[o,i], 0, scale ) )
// Block: 256 threads = 8 waves, 1 (M) x 8 (N); each wave owns a 32x32 tile
// built from 2x2 V_WMMA_F32_16X16X4_F32 tiles. A comes from LDS (shared by
// all 8 waves), B from L2-resident weights. Grid: (N/256, B/32) = (4, 32).
// ---------------------------------------------------------------------------
__global__ __launch_bounds__(256) void ucf_layer_kernel(
    const float* __restrict__ Hin,    // [B, N] previous activations
    const float* __restrict__ W,      // [N, N] this layer's weights, (o, i)
    float* __restrict__ Hout,         // [B, N]
    const int* __restrict__ gidx,     // [N] this layer's gather indices
    const float* __restrict__ act_scales,
    const int* __restrict__ tq_ptr,
    int layer) {
  __shared__ float As[BLK_M][LDS_STRIDE];   // 131,584 B of the 320 KB WGP LDS

  const int lane  = threadIdx.x & 31;
  const int waveN = threadIdx.x >> 5;     // 0..7
  const int lm = lane & 15;               // M (A) / N (B,C,D) index
  const int lh = lane >> 4;               // K half-select: K = 2*lh + {0,1}

  const int rowBase = blockIdx.y * BLK_M;          // global row base
  const int n0 = blockIdx.x * 256 + waveN * 32;    // global col base of tile

  // ---- Phase 1: gather this block's rows through idx into LDS -------------
  {
    const float* hrow = Hin + (size_t)rowBase * N_DIM;
    for (int e = threadIdx.x; e < BLK_M * N_DIM; e += 256) {
      int r = e >> 10;                   // N_DIM == 1024
      int j = e & (N_DIM - 1);
      As[r][j] = hrow[(size_t)r * N_DIM + gidx[j]];
    }
  }
  __syncthreads();

  // ---- Phase 2: WMMA GEMM, A from LDS, B from global (L2-hot) -------------
  const float* aBase0 = &As[lm][2 * lh];          // rows lm      (tiles c0x)
  const float* aBase1 = &As[16 + lm][2 * lh];     // rows 16+lm   (tiles c1x)
  const float* bRow0  = W + (size_t)(n0 + lm)      * N_DIM + 2 * lh;
  const float* bRow1  = W + (size_t)(n0 + 16 + lm) * N_DIM + 2 * lh;

  v8f c00 = {}; v8f c01 = {}; v8f c10 = {}; v8f c11 = {};

#pragma unroll 4
  for (int k0 = 0; k0 < N_DIM; k0 += 4) {
    v2f a0 = *(const v2f*)(aBase0 + k0);
    v2f a1 = *(const v2f*)(aBase1 + k0);
    v2f b0 = *(const v2f*)(bRow0 + k0);
    v2f b1 = *(const v2f*)(bRow1 + k0);
    // 8-arg form: (neg_a, A, neg_b, B, c_mod, C, reuse_a, reuse_b)
    c00 = __builtin_amdgcn_wmma_f32_16x16x4_f32(false, a0, false, b0,
                                                (short)0, c00, false, false);
    c01 = __builtin_amdgcn_wmma_f32_16x16x4_f32(false, a0, false, b1,
                                                (short)0, c01, false, false);
    c10 = __builtin_amdgcn_wmma_f32_16x16x4_f32(false, a1, false, b0,
                                                (short)0, c10, false, false);
    c11 = __builtin_amdgcn_wmma_f32_16x16x4_f32(false, a1, false, b1,
                                                (short)0, c11, false, false);
  }

  // ---- Phase 3: fused clip + fake-quant epilogue (forward value == q) -----
  const float scale  = act_scales[layer];
  const float tqf    = (float)(*tq_ptr);
  const float to_q   = tqf / scale;
  const float from_q = scale / tqf;

  // C/D layout: VGPR v <-> row (v + 8*lh), lane lm <-> column.
#pragma unroll
  for (int v = 0; v < 8; ++v) {
    int rA = rowBase + v + 8 * lh;      // rows for c00 / c01
    int rB = rA + 16;                   // rows for c10 / c11
    int cA = n0 + lm;                   // cols for c00 / c10
    int cB = cA + 16;                   // cols for c01 / c11

    float y;
    y = fminf(fmaxf(c00[v], 0.0f), scale);
    Hout[(size_t)rA * N_DIM + cA] = __builtin_rintf(y * to_q) * from_q;
    y = fminf(fmaxf(c01[v], 0.0f), scale);
    Hout[(size_t)rA * N_DIM + cB] = __builtin_rintf(y * to_q) * from_q;
    y = fminf(fmaxf(c10[v], 0.0f), scale);
    Hout[(size_t)rB * N_DIM + cA] = __builtin_rintf(y * to_q) * from_q;
    y = fminf(fmaxf(c11[v], 0.0f), scale);
    Hout[(size_t)rB * N_DIM + cB] = __builtin_rintf(y * to_q) * from_q;
  }
}

// ---------------------------------------------------------------------------
// Final output gather: out[b, t] = h[b, out_idx[t]]
// ---------------------------------------------------------------------------
__global__ __launch_bounds__(256) void ucf_gather_out(
    const float* __restrict__ h, const int* __restrict__ out_idx,
    float* __restrict__ out) {
  int t = blockIdx.x * 256 + threadIdx.x;
  if (t < B_DIM * N_OUT) {
    int b = t / N_OUT;
    int j = t - b * N_OUT;
    out[t] = h[(size_t)b * N_DIM + out_idx[j]];
  }
}

// ---------------------------------------------------------------------------
// Launch: 24 fused layer kernels ping-ponging h in d_ws, then output gather.
// All on `stream` (graph-capture safe).
// ---------------------------------------------------------------------------
extern "C" void kernel_launch(void* const* d_in, const int* in_sizes, int n_in,
                              void* d_out, int out_size, void* d_ws, size_t ws_size,
                              hipStream_t stream) {
  const float* x          = (const float*)d_in[0];   // [B, N]
  const float* weights    = (const float*)d_in[1];   // [L, N, N]
  const int*   gather_idx = (const int*)d_in[2];     // [L, N]
  const float* act_scales = (const float*)d_in[3];   // [L]
  const int*   out_idx    = (const int*)d_in[4];     // [N_OUT]
  const int*   tq_ptr     = (const int*)d_in[5];     // scalar Tq

  const size_t hElems = (size_t)B_DIM * N_DIM;
  float* ws0 = (float*)d_ws;                 // h ping
  float* ws1 = ws0 + hElems;                 // h pong

  const dim3 grid(N_DIM / 256, B_DIM / BLK_M);   // (4, 32)

  for (int l = 0; l < L_DIM; ++l) {
    const float* hin  = (l == 0) ? x : (((l - 1) & 1) ? ws1 : ws0);
    float*       hout = (l & 1) ? ws1 : ws0;

    ucf_layer_kernel<<<grid, 256, 0, stream>>>(
        hin, weights + (size_t)l * N_DIM * N_DIM, hout,
        gather_idx + (size_t)l * N_DIM, act_scales, tq_ptr, l);
  }

  const float* hFinal = ((L_DIM - 1) & 1) ? ws1 : ws0;   // L=24 -> ws1
  ucf_gather_out<<<(B_DIM * N_OUT + 255) / 256, 256, 0, stream>>>(
      hFinal, out_idx, (float*)d_out);
}